// Neuron1d_1829656068700
// MI455X (gfx1250) — compile-verified
//
#include <hip/hip_runtime.h>
#include <hip/hip_bf16.h>

// ---------------------------------------------------------------------------
// Neuron1d pipeline for MI455X (gfx1250, wave32):
//   similarity     = conv1d_same(x, w)        [fp32 WMMA 16x16x4 Toeplitz GEMM,
//                                              fused level-1 radix histogram]
//   extrema        = topk_abs(similarity, k)  [3-level radix select]
//   reconstruction = conv1d_same(extrema, w)  [same WMMA kernel, no hist]
// d_out = [reconstruction | similarity | extrema], each 64*262144 fp32.
// ---------------------------------------------------------------------------

typedef __attribute__((ext_vector_type(2))) float v2f;
typedef __attribute__((ext_vector_type(8))) float v8f;
typedef unsigned int uint32;

#define BATCH     64
#define LROW      262144
#define NTOT      (BATCH * LROW)
#define KSZ       257
#define PADK      128
#define WG_THR    256          // 8 waves of 32
#define TILE_WAVE 256          // outputs per wave (16x16 D tile, m + 16n)
#define TILE_WG   2048         // outputs per workgroup (8 waves)
#define NCHUNK    68           // reduction: q in [-128,143], K=4 per WMMA
#define WIN       (TILE_WG + 256)  // 2304-float x window per workgroup
#define WZLEN     304          // zero-padded weights: wz[i] = w[i-16]

#define NB_HI  2048
#define NB_MID 2048
#define NB_LO  512
#define HIST_WGS 32            // blocks per batch row for mid/lo histogram passes

// ---------------------------------------------------------------------------
// conv1d (cross-correlation, same padding) via V_WMMA_F32_16X16X4_F32.
// D[m][n] = sum_q A[m][q] * B[q][n],  A[m][q] = w[q+128-m] (0 outside),
// B[q][n] = x[l0w + 16n + q]  =>  out[l0w + m + 16n] = sum_j w[j] x[l + j - 128].
// When HIST: also accumulate level-1 radix histogram (bits [30:20] of |out|)
// straight from the accumulator registers -> saves a full 67 MB re-read.
// ---------------------------------------------------------------------------
template <bool HIST>
__global__ __launch_bounds__(WG_THR)
void conv1d_wmma(const float* __restrict__ xin, const float* __restrict__ w,
                 float* __restrict__ out, uint32* __restrict__ H1)
{
    __shared__ float xs[WIN];
    __shared__ float wz[WZLEN];
    __shared__ uint32 h[HIST ? NB_HI : 1];

    const int tid = threadIdx.x;
    const int b   = blockIdx.y;
    const int L0  = blockIdx.x * TILE_WG;
    const float* row = xin + (size_t)b * LROW;

    if (HIST) {
        for (int i = tid; i < NB_HI; i += WG_THR) h[i] = 0u;
    }
    for (int i = tid; i < WZLEN; i += WG_THR) {
        int j = i - 16;
        wz[i] = (j >= 0 && j < KSZ) ? w[j] : 0.0f;
    }
    for (int i = tid; i < WIN; i += WG_THR) {   // window [L0-128, L0+2176)
        int gl = L0 - PADK + i;
        xs[i] = (gl >= 0 && gl < LROW) ? row[gl] : 0.0f;
    }
    __syncthreads();

    const int wave  = tid >> 5;
    const int lane  = tid & 31;
    const int mn    = lane & 15;          // m for A rows / n for B cols
    const int kb    = (lane >> 4) << 1;   // K pair: lanes 0-15 -> K0/K1, 16-31 -> K2/K3
    const int wbase = wave * TILE_WAVE;

    v8f acc = {};
    for (int t = 0; t < NCHUNK; ++t) {
        const int q0 = -PADK + 4 * t;
        v2f a, bf;
        const int wi = q0 + kb - mn + 144;            // wz index of w[q+128-m]
        a[0] = wz[wi];
        a[1] = wz[wi + 1];
        const int xi = wbase + 16 * mn + q0 + kb + PADK;
        bf[0] = xs[xi];
        bf[1] = xs[xi + 1];
        acc = __builtin_amdgcn_wmma_f32_16x16x4_f32(false, a, false, bf,
                                                    (short)0, acc, false, false);
    }

    // D: VGPR r <-> M = r (lanes 0-15) / r+8 (lanes 16-31); N = lane&15.
    const int hi = lane >> 4;
    float* op = out + (size_t)b * LROW + L0 + wbase + 16 * mn + 8 * hi;
    *(float4*)(op)     = make_float4(acc[0], acc[1], acc[2], acc[3]);
    *(float4*)(op + 4) = make_float4(acc[4], acc[5], acc[6], acc[7]);

    if (HIST) {
        #pragma unroll
        for (int r = 0; r < 8; ++r) {
            uint32 u = __float_as_uint(acc[r]) & 0x7fffffffu;
            atomicAdd(&h[u >> 20], 1u);
        }
        __syncthreads();
        for (int i = tid; i < NB_HI; i += WG_THR)
            if (h[i]) atomicAdd(&H1[b * NB_HI + i], h[i]);
    }
}

// ---------------------------------------------------------------------------
// Radix select on u = bits(|x|) (order-preserving for non-negative floats).
// Level 1: bits [30:20] (fused above), level 2: [19:9], level 3: [8:0].
// ---------------------------------------------------------------------------
__global__ void zero_ws_kernel(uint32* p, int n) {
    int i = blockIdx.x * blockDim.x + threadIdx.x;
    if (i < n) p[i] = 0u;
}

__global__ void scan_hi(const uint32* __restrict__ H1, uint32* __restrict__ ST,
                        const int* __restrict__ kp)
{
    if (threadIdx.x != 0) return;
    const int b = blockIdx.x;
    const uint32 k = (uint32)(*kp);
    const uint32* h = H1 + b * NB_HI;
    uint32 cnt = 0; int bin = 0;
    for (int i = NB_HI - 1; i >= 0; --i) {
        if (cnt + h[i] >= k) { bin = i; break; }
        cnt += h[i];
    }
    ST[b * 8 + 0] = (uint32)bin << 20;  // prefix
    ST[b * 8 + 1] = cnt;                // # strictly above prefix range
}

__global__ __launch_bounds__(WG_THR)
void hist_mid(const float* __restrict__ sim, const uint32* __restrict__ ST,
              uint32* __restrict__ H2)
{
    __shared__ uint32 h[NB_MID];
    const int tid = threadIdx.x;
    for (int i = tid; i < NB_MID; i += WG_THR) h[i] = 0u;
    __syncthreads();
    const int b = blockIdx.y;
    const uint32 pref_hi = ST[b * 8 + 0] >> 20;
    const float4* row = (const float4*)(sim + (size_t)b * LROW);
    const int per4 = (LROW / 4) / HIST_WGS;
    const int start = blockIdx.x * per4;
    for (int i = start + tid; i < start + per4; i += WG_THR) {
        float4 v = row[i];
        uint32 u0 = __float_as_uint(v.x) & 0x7fffffffu;
        uint32 u1 = __float_as_uint(v.y) & 0x7fffffffu;
        uint32 u2 = __float_as_uint(v.z) & 0x7fffffffu;
        uint32 u3 = __float_as_uint(v.w) & 0x7fffffffu;
        if ((u0 >> 20) == pref_hi) atomicAdd(&h[(u0 >> 9) & 0x7ffu], 1u);
        if ((u1 >> 20) == pref_hi) atomicAdd(&h[(u1 >> 9) & 0x7ffu], 1u);
        if ((u2 >> 20) == pref_hi) atomicAdd(&h[(u2 >> 9) & 0x7ffu], 1u);
        if ((u3 >> 20) == pref_hi) atomicAdd(&h[(u3 >> 9) & 0x7ffu], 1u);
    }
    __syncthreads();
    for (int i = tid; i < NB_MID; i += WG_THR)
        if (h[i]) atomicAdd(&H2[b * NB_MID + i], h[i]);
}

__global__ void scan_mid(const uint32* __restrict__ H2, uint32* __restrict__ ST,
                         const int* __restrict__ kp)
{
    if (threadIdx.x != 0) return;
    const int b = blockIdx.x;
    const uint32 k = (uint32)(*kp);
    const uint32* h = H2 + b * NB_MID;
    uint32 cnt = ST[b * 8 + 1]; int bin = 0;
    for (int i = NB_MID - 1; i >= 0; --i) {
        if (cnt + h[i] >= k) { bin = i; break; }
        cnt += h[i];
    }
    ST[b * 8 + 0] |= (uint32)bin << 9;
    ST[b * 8 + 1] = cnt;
}

__global__ __launch_bounds__(WG_THR)
void hist_lo(const float* __restrict__ sim, const uint32* __restrict__ ST,
             uint32* __restrict__ H3)
{
    __shared__ uint32 h[NB_LO];
    const int tid = threadIdx.x;
    for (int i = tid; i < NB_LO; i += WG_THR) h[i] = 0u;
    __syncthreads();
    const int b = blockIdx.y;
    const uint32 pref22 = ST[b * 8 + 0] >> 9;
    const float4* row = (const float4*)(sim + (size_t)b * LROW);
    const int per4 = (LROW / 4) / HIST_WGS;
    const int start = blockIdx.x * per4;
    for (int i = start + tid; i < start + per4; i += WG_THR) {
        float4 v = row[i];
        uint32 u0 = __float_as_uint(v.x) & 0x7fffffffu;
        uint32 u1 = __float_as_uint(v.y) & 0x7fffffffu;
        uint32 u2 = __float_as_uint(v.z) & 0x7fffffffu;
        uint32 u3 = __float_as_uint(v.w) & 0x7fffffffu;
        if ((u0 >> 9) == pref22) atomicAdd(&h[u0 & 0x1ffu], 1u);
        if ((u1 >> 9) == pref22) atomicAdd(&h[u1 & 0x1ffu], 1u);
        if ((u2 >> 9) == pref22) atomicAdd(&h[u2 & 0x1ffu], 1u);
        if ((u3 >> 9) == pref22) atomicAdd(&h[u3 & 0x1ffu], 1u);
    }
    __syncthreads();
    for (int i = tid; i < NB_LO; i += WG_THR)
        if (h[i]) atomicAdd(&H3[b * NB_LO + i], h[i]);
}

__global__ void scan_lo(const uint32* __restrict__ H3, uint32* __restrict__ ST,
                        const int* __restrict__ kp)
{
    if (threadIdx.x != 0) return;
    const int b = blockIdx.x;
    const uint32 k = (uint32)(*kp);
    const uint32* h = H3 + b * NB_LO;
    uint32 cnt = ST[b * 8 + 1]; int bin = 0;
    for (int i = NB_LO - 1; i >= 0; --i) {
        if (cnt + h[i] >= k) { bin = i; break; }
        cnt += h[i];
    }
    ST[b * 8 + 2] = ST[b * 8 + 0] | (uint32)bin;  // exact threshold key T
    ST[b * 8 + 3] = k - cnt;                      // tie quota
    // ST[b*8+4] (tie counter) was zeroed by zero_ws_kernel this launch.
}

__global__ __launch_bounds__(WG_THR)
void select_scatter(const float* __restrict__ sim, float* __restrict__ ext,
                    uint32* __restrict__ ST)
{
    const int b = blockIdx.y;
    const uint32 T        = ST[b * 8 + 2];
    const uint32 tie_need = ST[b * 8 + 3];
    uint32* tie_ctr = &ST[b * 8 + 4];
    const float4* row = (const float4*)(sim + (size_t)b * LROW);
    float4* orow = (float4*)(ext + (size_t)b * LROW);
    const int per4 = (LROW / 4) / gridDim.x;
    const int start = blockIdx.x * per4;
    for (int i = start + threadIdx.x; i < start + per4; i += WG_THR) {
        float4 v = row[i];
        float4 o = make_float4(0.f, 0.f, 0.f, 0.f);
        #pragma unroll
        for (int c = 0; c < 4; ++c) {
            float  val = (&v.x)[c];
            uint32 u = __float_as_uint(val) & 0x7fffffffu;
            if (u > T) {
                (&o.x)[c] = val;
            } else if (u == T) {
                if (atomicAdd(tie_ctr, 1u) < tie_need) (&o.x)[c] = val;
            }
        }
        orow[i] = o;
    }
}

// ---------------------------------------------------------------------------
// Host launcher. Workspace layout (uint32):
//   H1[64*2048] | H2[64*2048] | H3[64*512] | STATE[64*8]  ~= 1.16 MB
// ---------------------------------------------------------------------------
extern "C" void kernel_launch(void* const* d_in, const int* in_sizes, int n_in,
                              void* d_out, int out_size, void* d_ws, size_t ws_size,
                              hipStream_t stream)
{
    (void)in_sizes; (void)n_in; (void)out_size; (void)ws_size;
    const float* x  = (const float*)d_in[0];
    const float* w  = (const float*)d_in[1];
    const int*   kp = (const int*)d_in[2];

    float* recon = (float*)d_out;
    float* sim   = (float*)d_out + NTOT;
    float* ext   = (float*)d_out + 2 * (size_t)NTOT;

    uint32* H1 = (uint32*)d_ws;
    uint32* H2 = H1 + BATCH * NB_HI;
    uint32* H3 = H2 + BATCH * NB_MID;
    uint32* ST = H3 + BATCH * NB_LO;
    const int wsn = BATCH * (NB_HI + NB_MID + NB_LO + 8);

    zero_ws_kernel<<<(wsn + 255) / 256, 256, 0, stream>>>(H1, wsn);

    dim3 cgrid(LROW / TILE_WG, BATCH);      // 128 x 64
    conv1d_wmma<true><<<cgrid, WG_THR, 0, stream>>>(x, w, sim, H1);

    dim3 hgrid(HIST_WGS, BATCH);            // 32 x 64
    scan_hi <<<BATCH, 32, 0, stream>>>(H1, ST, kp);
    hist_mid<<<hgrid, WG_THR, 0, stream>>>(sim, ST, H2);
    scan_mid<<<BATCH, 32, 0, stream>>>(H2, ST, kp);
    hist_lo <<<hgrid, WG_THR, 0, stream>>>(sim, ST, H3);
    scan_lo <<<BATCH, 32, 0, stream>>>(H3, ST, kp);

    dim3 sgrid(LROW / TILE_WG, BATCH);
    select_scatter<<<sgrid, WG_THR, 0, stream>>>(sim, ext, ST);

    conv1d_wmma<false><<<cgrid, WG_THR, 0, stream>>>(ext, w, recon, nullptr);
}